// SpamClassifierLstmCellPos_56736517980927
// MI455X (gfx1250) — compile-verified
//
#include <hip/hip_runtime.h>
#include <hip/hip_bf16.h>
#include <math.h>

#define VOCAB 32000
#define EMB   512
#define HID   512
#define NTAG  36
#define SEQ   512
#define G4H   (4*HID)      // 2048 gate rows
#define NBLK  32           // persistent blocks in recurrent phase

typedef float v2f __attribute__((ext_vector_type(2)));
typedef float v8f __attribute__((ext_vector_type(8)));

// ---------------------------------------------------------------------------
// Kernel 1: counting-sort timesteps by tag (single block, LDS atomics).
// order[grp_off[g] .. grp_off[g+1]) = timesteps whose tag == g.
// ---------------------------------------------------------------------------
__global__ void group_sort_kernel(const int* __restrict__ tag_ids,
                                  int* __restrict__ order,
                                  int* __restrict__ grp_off) {
  __shared__ int s_cnt[NTAG];
  __shared__ int s_off[NTAG + 1];
  int t = threadIdx.x;                       // 512 threads, one per timestep
  if (t < NTAG) s_cnt[t] = 0;
  __syncthreads();
  int tag = tag_ids[t];
  int rank = atomicAdd(&s_cnt[tag], 1);
  __syncthreads();
  if (t == 0) {
    int acc = 0;
    for (int g = 0; g < NTAG; ++g) { s_off[g] = acc; acc += s_cnt[g]; }
    s_off[NTAG] = acc;
  }
  __syncthreads();
  order[s_off[tag] + rank] = t;
  if (t <= NTAG) grp_off[t] = s_off[t];
}

// ---------------------------------------------------------------------------
// Kernel 2: embedding gather X[t][:] = emb[x[t]][:]; also init h buffer 0.
// ---------------------------------------------------------------------------
__global__ void embed_gather_kernel(const int* __restrict__ x,
                                    const float* __restrict__ emb,
                                    float* __restrict__ X,
                                    const float* __restrict__ h0,
                                    float* __restrict__ hbuf) {
  int t = blockIdx.x;
  int tok = x[t];
  for (int d = threadIdx.x; d < EMB; d += blockDim.x)
    X[(size_t)t * EMB + d] = emb[(size_t)tok * EMB + d];
  if (t == 0)
    for (int d = threadIdx.x; d < HID; d += blockDim.x)
      hbuf[d] = h0[d];
}

// ---------------------------------------------------------------------------
// Kernel 3: tag-grouped input projection via WMMA f32 16x16x4.
// Per tag g: PRE[t][row] = (W_ih[g] @ x_t)[row] + b_ih[g][row] + b_hh[g][row]
// for every timestep t in tag group g. grid = (16 row-tile groups, 36 tags),
// 256 threads = 8 waves; wave w handles row tile blockIdx.x*8+w (16 rows).
// A tile layout (16x4 f32, ISA 7.12.2): lane m=lane&15, a.x=K(lane<16?0:2),
// a.y = next K. B tile (4x16): lanes 0-15 hold K=0,1; lanes 16-31 K=2,3,
// n = lane&15. C/D: 8 VGPRs, M = r + (lane>=16)*8, N = lane&15.
// ---------------------------------------------------------------------------
__global__ void input_proj_wmma_kernel(const float* __restrict__ Wih,
                                       const float* __restrict__ bih,
                                       const float* __restrict__ bhh,
                                       const float* __restrict__ X,
                                       const int* __restrict__ order,
                                       const int* __restrict__ grp_off,
                                       float* __restrict__ PRE) {
  const int g   = blockIdx.y;                 // tag
  const int off = grp_off[g];
  const int ng  = grp_off[g + 1] - off;       // timesteps with this tag
  if (ng <= 0) return;

  const int wave = threadIdx.x >> 5;
  const int lane = threadIdx.x & 31;
  const int rt   = blockIdx.x * 8 + wave;     // row tile 0..127
  const int rowbase = rt * 16;
  const int m     = lane & 15;
  const int khalf = (lane >> 4) * 2;          // 0 or 2

  const float* Wg = Wih + (size_t)g * G4H * EMB;             // [2048][512]
  const float* wrow = Wg + (size_t)(rowbase + m) * EMB + khalf;
  const float* bih_g = bih + (size_t)g * G4H;
  const float* bhh_g = bhh + (size_t)g * G4H;

  for (int ct = 0; ct * 16 < ng; ++ct) {
    const int col  = ct * 16 + (lane & 15);   // column (timestep slot) for N=lane&15
    const bool valid = (col < ng);
    const int tcol = order[off + (valid ? col : 0)];
    const float* xrow = X + (size_t)tcol * EMB + khalf;

    v8f c = {0.f, 0.f, 0.f, 0.f, 0.f, 0.f, 0.f, 0.f};
    #pragma unroll 8
    for (int k = 0; k < EMB; k += 4) {
      v2f a; a.x = wrow[k]; a.y = wrow[k + 1];
      v2f b; b.x = xrow[k]; b.y = xrow[k + 1];
      c = __builtin_amdgcn_wmma_f32_16x16x4_f32(
              /*neg_a=*/false, a, /*neg_b=*/false, b,
              /*c_mod=*/(short)0, c, /*reuse_a=*/false, /*reuse_b=*/false);
    }
    if (valid) {
      const int mhi = (lane >> 4) * 8;
      float* dst = PRE + (size_t)tcol * G4H;
      #pragma unroll
      for (int r = 0; r < 8; ++r) {
        const int row = rowbase + r + mhi;
        dst[row] = c[r] + bih_g[row] + bhh_g[row];
      }
    }
  }
}

// ---------------------------------------------------------------------------
// Kernel 4: persistent serial recurrence. 32 blocks x 256 threads.
// Block b owns j = b*16 .. b*16+15. Thread layout: tid = jl*16 + gate*4 + sub.
// Each gate row dot(512) split over 4 lanes (128 k each, float4 vectorized),
// reduced by wave shuffles. One device-wide barrier per timestep (monotonic
// atomic counter, s_sleep spin). While spinning, the next timestep's W_hh
// rows are prefetched (tag sequence is known) to hide L2 latency on the
// serial dependency chain. h double-buffered in hbuf[2][512]; c in registers.
// ---------------------------------------------------------------------------
__global__ void recurrent_kernel(const float* __restrict__ Whh,
                                 const float* __restrict__ PRE,
                                 const int* __restrict__ tag_ids,
                                 const float* __restrict__ c0,
                                 float* __restrict__ hbuf,      // [2][HID]
                                 unsigned int* __restrict__ bar,
                                 const float* __restrict__ Wfc,
                                 const float* __restrict__ bfc,
                                 float* __restrict__ out) {
  __shared__ float sh[HID];
  __shared__ float racc[256];

  const int tid   = threadIdx.x;
  const int jl    = tid >> 4;            // 0..15
  const int inner = tid & 15;
  const int gate  = (tid >> 2) & 3;      // 0:i 1:f 2:g 3:o
  const int sub   = tid & 3;             // k-strip
  const int j     = blockIdx.x * 16 + jl;
  const int grow  = gate * HID + j;      // gate row in [0,2048)
  const bool owner = (inner == 0);

  float c = owner ? c0[j] : 0.f;

  // warm caches for step 0 weights
  {
    const int tag0 = tag_ids[0];
    const char* w0 = (const char*)(Whh + ((size_t)tag0 * G4H + grow) * HID + sub * 128);
    #pragma unroll
    for (int l = 0; l < 4; ++l) __builtin_prefetch(w0 + l * 128, 0, 1);
  }

  for (int t = 0; t < SEQ; ++t) {
    // stage current h into LDS
    const float* hsrc = hbuf + (size_t)(t & 1) * HID;
    for (int d = tid; d < HID; d += 256) sh[d] = hsrc[d];
    __syncthreads();

    const int tag = tag_ids[t];
    const float4* w4 = (const float4*)(Whh + ((size_t)tag * G4H + grow) * HID + sub * 128);
    const float4* h4 = (const float4*)(sh + sub * 128);
    float p = 0.f;
    #pragma unroll 8
    for (int k = 0; k < 32; ++k) {
      const float4 a = w4[k];
      const float4 b = h4[k];
      p = fmaf(a.x, b.x, p);
      p = fmaf(a.y, b.y, p);
      p = fmaf(a.z, b.z, p);
      p = fmaf(a.w, b.w, p);
    }
    // reduce the 4 k-strips (lanes tid^1, tid^2 within wave32)
    p += __shfl_xor(p, 1, 32);
    p += __shfl_xor(p, 2, 32);
    const float gv = p + PRE[(size_t)t * G4H + grow];

    // gather i,f,g,o to the owner lane (lane 0 or 16 of the wave)
    const float gi = gv;
    const float gf = __shfl_down(gv, 4, 32);
    const float gg = __shfl_down(gv, 8, 32);
    const float go = __shfl_down(gv, 12, 32);

    if (owner) {
      const float i_ = 1.f / (1.f + __expf(-gi));
      const float f_ = 1.f / (1.f + __expf(-gf));
      const float g_ = tanhf(gg);
      const float o_ = 1.f / (1.f + __expf(-go));
      c = f_ * c + i_ * g_;
      const float hn = o_ * tanhf(c);
      hbuf[(size_t)((t + 1) & 1) * HID + j] = hn;
      if (t == SEQ - 1) { out[1 + j] = hn; out[1 + HID + j] = c; }
    }

    // prefetch next step's weight strip; overlaps with the barrier spin
    if (t + 1 < SEQ) {
      const int ntag = tag_ids[t + 1];
      const char* wn = (const char*)(Whh + ((size_t)ntag * G4H + grow) * HID + sub * 128);
      #pragma unroll
      for (int l = 0; l < 4; ++l) __builtin_prefetch(wn + l * 128, 0, 1);
    }

    // publish h, then device-wide barrier (monotonic counter, no reset race)
    __threadfence();
    __syncthreads();
    if (tid == 0) {
      __hip_atomic_fetch_add(bar, 1u, __ATOMIC_ACQ_REL, __HIP_MEMORY_SCOPE_AGENT);
      const unsigned target = (unsigned)NBLK * (unsigned)(t + 1);
      while (__hip_atomic_load(bar, __ATOMIC_ACQUIRE, __HIP_MEMORY_SCOPE_AGENT) < target)
        __builtin_amdgcn_s_sleep(1);
    }
    __syncthreads();
  }

  // final FC + sigmoid on block 0 (h_final lives in hbuf[SEQ&1] == hbuf[0])
  if (blockIdx.x == 0) {
    const float* hf = hbuf + (size_t)(SEQ & 1) * HID;
    float p = 0.f;
    for (int d = tid; d < HID; d += 256) p = fmaf(hf[d], Wfc[d], p);
    racc[tid] = p;
    __syncthreads();
    for (int s = 128; s > 0; s >>= 1) {
      if (tid < s) racc[tid] += racc[tid + s];
      __syncthreads();
    }
    if (tid == 0) out[0] = 1.f / (1.f + __expf(-(racc[0] + bfc[0])));
  }
}

// ---------------------------------------------------------------------------
// Launch
// ---------------------------------------------------------------------------
extern "C" void kernel_launch(void* const* d_in, const int* in_sizes, int n_in,
                              void* d_out, int out_size, void* d_ws, size_t ws_size,
                              hipStream_t stream) {
  (void)in_sizes; (void)n_in; (void)out_size; (void)ws_size;
  const int*   x       = (const int*)  d_in[0];
  const int*   tag_ids = (const int*)  d_in[1];
  const float* h0      = (const float*)d_in[2];
  const float* c0      = (const float*)d_in[3];
  const float* emb     = (const float*)d_in[4];
  const float* Wih     = (const float*)d_in[5];
  const float* Whh     = (const float*)d_in[6];
  const float* bih     = (const float*)d_in[7];
  const float* bhh     = (const float*)d_in[8];
  const float* Wfc     = (const float*)d_in[9];
  const float* bfc     = (const float*)d_in[10];
  float* out = (float*)d_out;

  // workspace layout
  char* ws = (char*)d_ws;
  float* PRE  = (float*)(ws);                       // 512*2048*4 = 4 MB
  float* X    = (float*)(ws + 4194304);             // 512*512*4  = 1 MB
  float* hbuf = (float*)(ws + 5242880);             // 2*512*4
  int*   order   = (int*)(ws + 5246976);            // 512*4
  int*   grp_off = (int*)(ws + 5249024);            // 37*4
  unsigned int* bar = (unsigned int*)(ws + 5249280);

  // barrier counter must start at 0 every launch (capture-safe memset node)
  hipMemsetAsync(ws + 5249280, 0, 256, stream);

  group_sort_kernel<<<1, 512, 0, stream>>>(tag_ids, order, grp_off);
  embed_gather_kernel<<<SEQ, 256, 0, stream>>>(x, emb, X, h0, hbuf);
  input_proj_wmma_kernel<<<dim3(16, NTAG), 256, 0, stream>>>(
      Wih, bih, bhh, X, order, grp_off, PRE);
  recurrent_kernel<<<NBLK, 256, 0, stream>>>(
      Whh, PRE, tag_ids, c0, hbuf, bar, Wfc, bfc, out);
}